// CNNFocusing_31928786878912
// MI455X (gfx1250) — compile-verified
//
#include <hip/hip_runtime.h>

typedef __attribute__((ext_vector_type(2))) float v2f;
typedef __attribute__((ext_vector_type(8))) float v8f;

#define NG   72          // landmark groups
#define NHP  10          // hidden per landmark
#define NHD  1536        // 6 * 16 * 16
#define NB   64          // batch

// Async global->LDS copy of 16B per active lane (CDNA5 GLOBAL_LOAD_ASYNC_TO_LDS_B128,
// tracked by ASYNCcnt).  lds_addr is the low-32-bit LDS byte address (flat-aperture
// rule: the low 32 address bits are used).  Pattern: issue -> s_wait_asynccnt 0 -> barrier.
__device__ __forceinline__ void async_copy_b128(void* lds_dst, const void* gsrc) {
  asm volatile("global_load_async_to_lds_b128 %0, %1, off"
               :
               : "v"((unsigned)(unsigned long long)lds_dst), "v"(gsrc)
               : "memory");
}
__device__ __forceinline__ void async_wait_all() {
  asm volatile("s_wait_asynccnt 0x0" ::: "memory");
}

// ---------------------------------------------------------------------------
// Kernel 1: grouped conv1 (3in/3out per group, 3x3, edge pad) + ReLU + 2x2 pool.
// One block = one (batch, group): stage the 3 input planes (3x32x32 = 12 KB)
// into LDS with async-to-LDS copies, then each thread computes one pooled
// pixel for all 3 out-channels of the group from LDS.
// ---------------------------------------------------------------------------
__global__ void conv1_pool_kernel(const float* __restrict__ x,
                                  const float* __restrict__ w,
                                  const float* __restrict__ bias,
                                  float* __restrict__ h1p) {
  __shared__ float xs[3 * 1024];           // 12 KB
  const int blk = blockIdx.x;
  const int b = blk / NG;
  const int g = blk % NG;
  const int t = threadIdx.x;               // 0..255
  const int ox = t & 15;
  const int oy = t >> 4;

  const float* xb = x + ((long)b * 216 + (long)g * 3) * 1024;

  // Stage 3072 floats = 768 x 16B: 3 chunks of 16B per thread.
  #pragma unroll
  for (int r = 0; r < 3; ++r) {
    const int idx = t + 256 * r;           // float4 index 0..767
    async_copy_b128(&xs[idx * 4], (const float4*)xb + idx);
  }
  async_wait_all();
  __syncthreads();

  // 3 out-channels of this group reuse the staged planes.
  #pragma unroll
  for (int j = 0; j < 3; ++j) {
    const int c = g * 3 + j;
    float wr[27];
    const float* wp = w + c * 27;          // uniform per block -> scalar loads
    #pragma unroll
    for (int i = 0; i < 27; ++i) wr[i] = wp[i];
    const float bv = bias[c];

    float m = -3.402823466e38f;
    #pragma unroll
    for (int dy = 0; dy < 2; ++dy) {
      #pragma unroll
      for (int dx = 0; dx < 2; ++dx) {
        const int y0 = oy * 2 + dy;
        const int x0 = ox * 2 + dx;
        float acc = bv;
        #pragma unroll
        for (int i = 0; i < 3; ++i) {
          #pragma unroll
          for (int ky = 0; ky < 3; ++ky) {
            int yy = y0 + ky - 1;
            yy = yy < 0 ? 0 : (yy > 31 ? 31 : yy);     // edge padding
            #pragma unroll
            for (int kx = 0; kx < 3; ++kx) {
              int xx = x0 + kx - 1;
              xx = xx < 0 ? 0 : (xx > 31 ? 31 : xx);
              acc = fmaf(xs[i * 1024 + yy * 32 + xx], wr[i * 9 + ky * 3 + kx], acc);
            }
          }
        }
        acc = fmaxf(acc, 0.0f);
        m = fmaxf(m, acc);
      }
    }
    h1p[((long)b * 216 + c) * 256 + t] = m;
  }
}

// ---------------------------------------------------------------------------
// Kernel 2: grouped conv2 (3in/6out per group, 3x3 edge pad) + ReLU on 16x16.
// One block = one (batch, group): stage 3x16x16 = 3 KB into LDS (async),
// each thread computes one pixel for all 6 out-channels.
// ---------------------------------------------------------------------------
__global__ void conv2_kernel(const float* __restrict__ h1p,
                             const float* __restrict__ w,
                             const float* __restrict__ bias,
                             float* __restrict__ h2) {
  __shared__ float hs[3 * 256];            // 3 KB
  const int blk = blockIdx.x;
  const int b = blk / NG;
  const int g = blk % NG;
  const int t = threadIdx.x;
  const int ox = t & 15;
  const int oy = t >> 4;

  const float* xb = h1p + ((long)b * 216 + (long)g * 3) * 256;

  // 768 floats = 192 x 16B: first 6 waves stage one chunk each.
  if (t < 192) async_copy_b128(&hs[t * 4], (const float4*)xb + t);
  async_wait_all();
  __syncthreads();

  #pragma unroll
  for (int j = 0; j < 6; ++j) {
    const int o = g * 6 + j;
    float wr[27];
    const float* wp = w + o * 27;
    #pragma unroll
    for (int i = 0; i < 27; ++i) wr[i] = wp[i];

    float acc = bias[o];
    #pragma unroll
    for (int i = 0; i < 3; ++i) {
      #pragma unroll
      for (int ky = 0; ky < 3; ++ky) {
        int yy = oy + ky - 1;
        yy = yy < 0 ? 0 : (yy > 15 ? 15 : yy);
        #pragma unroll
        for (int kx = 0; kx < 3; ++kx) {
          int xx = ox + kx - 1;
          xx = xx < 0 ? 0 : (xx > 15 ? 15 : xx);
          acc = fmaf(hs[i * 256 + yy * 16 + xx], wr[i * 9 + ky * 3 + kx], acc);
        }
      }
    }
    h2[((long)b * 432 + o) * 256 + t] = fmaxf(acc, 0.0f);
  }
}

// ---------------------------------------------------------------------------
// Kernel 3: block-diagonal FC1 via V_WMMA_F32_16X16X4_F32, then tiny FC2.
// One wave (32 lanes, EXEC all-1s) per (group g, batch tile of 16).
//   A (16x4, f32): lanes 0-15 hold M=0..15 K={k,k+1}; lanes 16-31 K={k+2,k+3}
//   B (4x16, f32): mirrored — lane%16 = column N, lane/16 selects K pair
//   C/D (16x16 f32, 8 VGPRs): VGPR r, lanes 0-15 -> M=r; lanes 16-31 -> M=r+8
// ---------------------------------------------------------------------------
__global__ void fc_wmma_kernel(const float* __restrict__ h2,
                               const float* __restrict__ w1,
                               const float* __restrict__ b1,
                               const float* __restrict__ w2,
                               const float* __restrict__ b2,
                               float* __restrict__ out) {
  __shared__ float zt[16][17];               // z tile [batch][hp], padded

  const int g    = blockIdx.x % NG;
  const int mt   = blockIdx.x / NG;          // 0..3 (batch tile)
  const int lane = threadIdx.x;              // 0..31
  const int nl   = lane & 15;
  const int hh   = lane >> 4;                // 0/1 -> K offset 0/2

  // A: activations hf[batch = mt*16 + nl, g, k], float2 along K
  const float* ap = h2 + ((long)(mt * 16 + nl)) * (432 * 256)
                       + (long)g * NHD + 2 * hh;
  // B: w1 diagonal block rows; rows >= NHP clamped (those D columns unused)
  const int wrow = (nl < NHP) ? nl : (NHP - 1);
  const float* bp = w1 + ((long)(g * NHP + wrow)) * (NG * NHD)
                       + (long)g * NHD + 2 * hh;

  // Prefetch the tiny FC2 weight row while the WMMA K-loop runs.
  const int bo = g * 2 + hh;
  const float* w2p = w2 + (long)bo * (NG * NHP) + (long)g * NHP;
  __builtin_prefetch(w2p, 0, 0);             // -> global_prefetch_b8

  v8f c = {};                                // zero accumulator
  #pragma unroll 4
  for (int k = 0; k < NHD; k += 4) {
    v2f a = *(const v2f*)(ap + k);
    v2f bb = *(const v2f*)(bp + k);
    c = __builtin_amdgcn_wmma_f32_16x16x4_f32(
            /*neg_a=*/false, a, /*neg_b=*/false, bb,
            /*c_mod=*/(short)0, c, /*reuse_a=*/false, /*reuse_b=*/false);
  }

  // Spill D to LDS: VGPR r on lane l holds z[M = r + 8*(l/16)][N = l%16]
  #pragma unroll
  for (int r = 0; r < 8; ++r) zt[r + 8 * hh][nl] = c[r];
  __syncthreads();

  // FC2: lane -> (batch = nl, output = hh); 10-term dot + biases
  float acc = b2[bo];
  const float* b1p = b1 + g * NHP;
  #pragma unroll
  for (int n = 0; n < NHP; ++n)
    acc = fmaf(zt[nl][n] + b1p[n], w2p[n], acc);

  out[((long)(mt * 16 + nl)) * (NG * 2) + bo] = acc;
}

// ---------------------------------------------------------------------------
extern "C" void kernel_launch(void* const* d_in, const int* in_sizes, int n_in,
                              void* d_out, int out_size, void* d_ws, size_t ws_size,
                              hipStream_t stream) {
  const float* x   = (const float*)d_in[0];
  const float* c1w = (const float*)d_in[1];
  const float* c1b = (const float*)d_in[2];
  const float* c2w = (const float*)d_in[3];
  const float* c2b = (const float*)d_in[4];
  const float* w1  = (const float*)d_in[5];
  const float* b1  = (const float*)d_in[6];
  const float* w2  = (const float*)d_in[7];
  const float* b2  = (const float*)d_in[8];
  float* out = (float*)d_out;

  float* h1p = (float*)d_ws;                       // [64,216,16,16] = 14.2 MB
  float* h2  = h1p + (long)NB * 216 * 256;         // [64,432,16,16] = 28.3 MB

  conv1_pool_kernel<<<NB * NG, 256, 0, stream>>>(x, c1w, c1b, h1p);
  conv2_kernel<<<NB * NG, 256, 0, stream>>>(h1p, c2w, c2b, h2);
  fc_wmma_kernel<<<NG * 4, 32, 0, stream>>>(h2, w1, b1, w2, b2, out);
}